// Star_Label_AM_27797028340035
// MI455X (gfx1250) — compile-verified
//
#include <hip/hip_runtime.h>

typedef __attribute__((ext_vector_type(2))) float v2f;
typedef __attribute__((ext_vector_type(8))) float v8f;

#define NB   512    // batch
#define SD   128    // seq len
#define DD   768    // hidden dim
#define NL   5      // num labels
#define EPSF 1e-8f

// ---------------------------------------------------------------------------
// 1) cls = embs[:,0,:]  (copy to output) + clamped row norms
// ---------------------------------------------------------------------------
__global__ void cls_norm_kernel(const float* __restrict__ embs,
                                float* __restrict__ cls,
                                float* __restrict__ norms) {
  __shared__ float red[256];
  const int i = blockIdx.x;
  const float* src = embs + (size_t)i * SD * DD;   // row i, seq position 0
  float ss = 0.0f;
  for (int d = threadIdx.x; d < DD; d += 256) {
    float v = src[d];
    cls[i * DD + d] = v;
    ss += v * v;
  }
  red[threadIdx.x] = ss;
  __syncthreads();
  for (int st = 128; st > 0; st >>= 1) {
    if (threadIdx.x < st) red[threadIdx.x] += red[threadIdx.x + st];
    __syncthreads();
  }
  if (threadIdx.x == 0) norms[i] = fmaxf(sqrtf(red[0]), EPSF);
}

// ---------------------------------------------------------------------------
// 2) label counts (as float) + star row inverse norms
// ---------------------------------------------------------------------------
__global__ void counts_star_kernel(const int* __restrict__ labels,
                                   const float* __restrict__ star,
                                   float* __restrict__ counts,
                                   float* __restrict__ star_inv) {
  const int t = threadIdx.x;
  if (t < NL) {
    float ss = 0.0f;
    const float* r = star + t * DD;
    for (int d = 0; d < DD; ++d) ss += r[d] * r[d];
    star_inv[t] = 1.0f / fmaxf(sqrtf(ss), EPSF);
  }
  if (t == 8) {
    int c0 = 0, c1 = 0, c2 = 0, c3 = 0, c4 = 0;
    for (int i = 0; i < NB; ++i) {
      int l = labels[i];
      c0 += (l == 0); c1 += (l == 1); c2 += (l == 2); c3 += (l == 3); c4 += (l == 4);
    }
    counts[0] = (float)c0; counts[1] = (float)c1; counts[2] = (float)c2;
    counts[3] = (float)c3; counts[4] = (float)c4;
  }
}

// ---------------------------------------------------------------------------
// 3) class sums of normalized rows: s[c][d] = sum_{i: label_i==c} cls[i][d]/norm[i]
// ---------------------------------------------------------------------------
__global__ void class_sum_kernel(const float* __restrict__ cls,
                                 const float* __restrict__ norms,
                                 const int* __restrict__ labels,
                                 float* __restrict__ s) {
  __shared__ int   sl[NB];
  __shared__ float si[NB];
  for (int j = threadIdx.x; j < NB; j += blockDim.x) {
    sl[j] = labels[j];
    si[j] = 1.0f / norms[j];
  }
  __syncthreads();
  const int d = blockIdx.x * blockDim.x + threadIdx.x;
  if (d >= DD) return;
  float a0 = 0, a1 = 0, a2 = 0, a3 = 0, a4 = 0;
  for (int i = 0; i < NB; ++i) {
    float nv = cls[i * DD + d] * si[i];
    int l = sl[i];
    a0 += (l == 0) ? nv : 0.0f;
    a1 += (l == 1) ? nv : 0.0f;
    a2 += (l == 2) ? nv : 0.0f;
    a3 += (l == 3) ? nv : 0.0f;
    a4 += (l == 4) ? nv : 0.0f;
  }
  s[0 * DD + d] = a0; s[1 * DD + d] = a1; s[2 * DD + d] = a2;
  s[3 * DD + d] = a3; s[4 * DD + d] = a4;
}

// ---------------------------------------------------------------------------
// 4) h = gelu(cls @ W_dense + b_dense)  — fp32 WMMA 16x16x4.
//    One wave per 16x16 output tile; 4 waves (4 N-tiles) per 128-thread block.
//    A frag: a[v] = A[lane&15][k + 2*(lane>>4) + v]       (16x4, 2 VGPRs)
//    B frag: b[v] = B[k + 2*(lane>>4) + v][lane&15]       (4x16, 2 VGPRs)
//    D:      acc[v] = D[v + 8*(lane>>4)][lane&15]         (16x16, 8 VGPRs)
// ---------------------------------------------------------------------------
__global__ void __launch_bounds__(128)
gemm_gelu_wmma(const float* __restrict__ A,      // [NB][DD]
               const float* __restrict__ W,      // [DD][DD], W[k][n]
               const float* __restrict__ bias,   // [DD]
               float* __restrict__ H) {          // [NB][DD]
  const int lane = threadIdx.x & 31;
  const int wave = threadIdx.x >> 5;
  const int m0 = blockIdx.x * 16;
  const int n0 = (blockIdx.y * 4 + wave) * 16;
  const int rc = lane & 15;        // M-row for A, N-col for B/D
  const int kh = lane >> 4;        // K-half select (0 or 1)

  const float* Ap = A + (size_t)(m0 + rc) * DD;
  const float* Bp = W + n0 + rc;

  v8f acc = {};
  for (int k = 0; k < DD; k += 4) {
    const int ka = k + 2 * kh;
    v2f a;
    a.x = Ap[ka];
    a.y = Ap[ka + 1];
    v2f b;
    b.x = Bp[(size_t)ka * DD];
    b.y = Bp[(size_t)(ka + 1) * DD];
    acc = __builtin_amdgcn_wmma_f32_16x16x4_f32(
        /*neg_a=*/false, a, /*neg_b=*/false, b,
        /*c_mod=*/(short)0, acc, /*reuse_a=*/false, /*reuse_b=*/false);
  }

  const int n = n0 + rc;
  const float bn = bias[n];
#pragma unroll
  for (int v = 0; v < 8; ++v) {
    const int m = m0 + v + 8 * kh;
    float x = acc[v] + bn;
    // exact GELU: 0.5*x*(1+erf(x/sqrt(2)))
    H[(size_t)m * DD + n] = 0.5f * x * (1.0f + erff(x * 0.70710678118654752f));
  }
}

// ---------------------------------------------------------------------------
// 5) logits = h @ W_out + b_out   (512x5, one thread per element)
// ---------------------------------------------------------------------------
__global__ void logits_kernel(const float* __restrict__ h,
                              const float* __restrict__ Wo,   // [DD][NL]
                              const float* __restrict__ bo,
                              float* __restrict__ logits) {
  const int idx = blockIdx.x * blockDim.x + threadIdx.x;
  if (idx >= NB * NL) return;
  const int i = idx / NL;
  const int l = idx % NL;
  float acc = bo[l];
  const float* hr = h + (size_t)i * DD;
  for (int d = 0; d < DD; ++d) acc += hr[d] * Wo[d * NL + l];
  logits[i * NL + l] = acc;
}

// ---------------------------------------------------------------------------
// 6) per-row contrastive / star terms:
//    per2[i] = (count_c - n_i . s_c) / count_c,   c = label_i
//    per3[i] = 1 - (n_i . star_c) * star_inv[c]
// ---------------------------------------------------------------------------
__global__ void per_row_kernel(const float* __restrict__ cls,
                               const float* __restrict__ norms,
                               const int* __restrict__ labels,
                               const float* __restrict__ s,
                               const float* __restrict__ counts,
                               const float* __restrict__ star,
                               const float* __restrict__ star_inv,
                               float* __restrict__ per2,
                               float* __restrict__ per3) {
  __shared__ float r1[256];
  __shared__ float r2[256];
  const int i = blockIdx.x;
  const int c = labels[i];
  const float inv = 1.0f / norms[i];
  float d1 = 0.0f, d2 = 0.0f;
  for (int d = threadIdx.x; d < DD; d += 256) {
    float nv = cls[i * DD + d] * inv;
    d1 += nv * s[c * DD + d];
    d2 += nv * star[c * DD + d];
  }
  r1[threadIdx.x] = d1;
  r2[threadIdx.x] = d2;
  __syncthreads();
  for (int st = 128; st > 0; st >>= 1) {
    if (threadIdx.x < st) {
      r1[threadIdx.x] += r1[threadIdx.x + st];
      r2[threadIdx.x] += r2[threadIdx.x + st];
    }
    __syncthreads();
  }
  if (threadIdx.x == 0) {
    const float cnt = counts[c];              // >= 1 (row i itself)
    per2[i] = (cnt - r1[0]) / cnt;
    per3[i] = 1.0f - r2[0] * star_inv[c];
  }
}

// ---------------------------------------------------------------------------
// 7) finalize: CE from log-softmax + mean reductions -> losses[3]
// ---------------------------------------------------------------------------
__global__ void finalize_kernel(const float* __restrict__ logits,
                                const int* __restrict__ labels,
                                const float* __restrict__ per2,
                                const float* __restrict__ per3,
                                float* __restrict__ losses) {
  __shared__ float r1[NB];
  __shared__ float r2[NB];
  __shared__ float r3[NB];
  const int i = threadIdx.x;     // 512 threads
  const float* lg = logits + i * NL;
  float mx = lg[0];
  for (int k = 1; k < NL; ++k) mx = fmaxf(mx, lg[k]);
  float se = 0.0f;
  for (int k = 0; k < NL; ++k) se += expf(lg[k] - mx);
  const float lp = lg[labels[i]] - mx - logf(se);
  r1[i] = -lp;
  r2[i] = per2[i];
  r3[i] = per3[i];
  __syncthreads();
  for (int st = NB / 2; st > 0; st >>= 1) {
    if (i < st) {
      r1[i] += r1[i + st];
      r2[i] += r2[i + st];
      r3[i] += r3[i + st];
    }
    __syncthreads();
  }
  if (i == 0) {
    const float invn = 1.0f / (float)NB;
    losses[0] = r1[0] * invn;
    losses[1] = 0.5f * r2[0] * invn;
    losses[2] = 0.5f * r3[0] * invn;
  }
}

// ---------------------------------------------------------------------------
extern "C" void kernel_launch(void* const* d_in, const int* in_sizes, int n_in,
                              void* d_out, int out_size, void* d_ws, size_t ws_size,
                              hipStream_t stream) {
  const float* embs    = (const float*)d_in[0];   // (512,128,768)
  const int*   labels  = (const int*)  d_in[1];   // (512,)
  const float* W_dense = (const float*)d_in[2];   // (768,768)
  const float* b_dense = (const float*)d_in[3];   // (768,)
  const float* W_out   = (const float*)d_in[4];   // (768,5)
  const float* b_out   = (const float*)d_in[5];   // (5,)
  const float* star    = (const float*)d_in[6];   // (5,768)

  float* out    = (float*)d_out;
  float* losses = out;                    // [3]
  float* logits = out + 3;                // [512*5]
  float* cls    = out + 3 + NB * NL;      // [512*768]

  float* ws       = (float*)d_ws;
  float* h        = ws;                   // 512*768
  float* norms    = h + NB * DD;          // 512
  float* ssum     = norms + NB;           // 5*768
  float* counts   = ssum + NL * DD;       // 5
  float* star_inv = counts + NL;          // 5
  float* per2     = star_inv + NL;        // 512
  float* per3     = per2 + NB;            // 512

  cls_norm_kernel<<<NB, 256, 0, stream>>>(embs, cls, norms);
  counts_star_kernel<<<1, 32, 0, stream>>>(labels, star, counts, star_inv);
  class_sum_kernel<<<(DD + 255) / 256, 256, 0, stream>>>(cls, norms, labels, ssum);
  gemm_gelu_wmma<<<dim3(NB / 16, (DD / 16) / 4), 128, 0, stream>>>(cls, W_dense, b_dense, h);
  logits_kernel<<<(NB * NL + 255) / 256, 256, 0, stream>>>(h, W_out, b_out, logits);
  per_row_kernel<<<NB, 256, 0, stream>>>(cls, norms, labels, ssum, counts, star,
                                         star_inv, per2, per3);
  finalize_kernel<<<1, NB, 0, stream>>>(logits, labels, per2, per3, losses);
}